// BoundaryUncertaintyLoss_18614388261343
// MI455X (gfx1250) — compile-verified
//
#include <hip/hip_runtime.h>

typedef __attribute__((ext_vector_type(2))) float v2f;
typedef __attribute__((ext_vector_type(8))) float v8f;

// Problem constants (match reference setup_inputs)
constexpr int B = 8, C = 19, H = 512, W = 512;
constexpr int HALO = 3;               // BOUNDARY_WIDTH
constexpr int TILE = 256;             // threads per block == strip width
constexpr int LROW = TILE + 2 * HALO; // 262 columns in LDS tile

// Sum the 32 lane values of a wave using V_WMMA_F32_16X16X4_F32.
// A = ones(16x4), B vgpr0 = lane value, B vgpr1 = 0  =>  D[m][n] = colsum(B)[n]
// and sum_n D[m][n] == sum of all lane values for every row m, independent of
// the exact lane->(K,N) striping of B's first VGPR.
__device__ __forceinline__ float wave_sum_wmma(float v) {
    v2f a; a.x = 1.0f; a.y = 1.0f;
    v2f b; b.x = v;    b.y = 0.0f;
    v8f c = {};
    v8f d = __builtin_amdgcn_wmma_f32_16x16x4_f32(
        /*neg_a=*/false, a, /*neg_b=*/false, b,
        /*c_mod=*/(short)0, c, /*reuse_a=*/false, /*reuse_b=*/false);
    float s = d[0];                       // lane L holds D[0 or 8][L%16]
    s += __shfl_xor(s, 8, 32);            // reduce 16 columns (stay in 16-group)
    s += __shfl_xor(s, 4, 32);
    s += __shfl_xor(s, 2, 32);
    s += __shfl_xor(s, 1, 32);
    return s;                             // every lane: wave total
}

__global__ __launch_bounds__(TILE)
void boundary_entropy_partial(const float* __restrict__ pred,
                              const long long* __restrict__ labels,
                              float* __restrict__ partial) {
    __shared__ unsigned char fg[7][LROW];
    __shared__ float wsum[TILE / 32];

    const int tid = threadIdx.x;
    const int x0  = blockIdx.x * TILE;
    const int y   = blockIdx.y;
    const int b   = blockIdx.z;

    // ---- load 7 x 262 foreground-flag tile (zero-padded like SAME conv) ----
    const long long* lab_b = labels + (long long)b * H * W;
    for (int i = tid; i < 7 * LROW; i += TILE) {
        int r  = i / LROW;            // 0..6  -> dy = r-3
        int cc = i - r * LROW;        // 0..261
        int gy = y + r - HALO;
        int gx = x0 + cc - HALO;
        unsigned char f = 0;
        if (gy >= 0 && gy < H && gx >= 0 && gx < W)
            f = (lab_b[(long long)gy * W + gx] != 0) ? 1 : 0;
        fg[r][cc] = f;
    }
    __syncthreads();

    // ---- boundary mask: background pixel within disk(3) of foreground ----
    const int lx = tid + HALO;
    bool center_bg = (fg[HALO][lx] == 0);
    bool near_fg = false;
    // per-dy half-width of disk radius 3: dy=-3..3 -> {0,2,2,3,2,2,0}
    constexpr int hw[7] = {0, 2, 2, 3, 2, 2, 0};
#pragma unroll
    for (int r = 0; r < 7; ++r) {
#pragma unroll
        for (int dx = -3; dx <= 3; ++dx) {
            if (dx >= -hw[r] && dx <= hw[r])
                near_fg = near_fg | (fg[r][lx + dx] != 0);
        }
    }
    const bool active = center_bg && near_fg;

    // ---- per-pixel softmax entropy (only if some lane in the wave needs it) ----
    float ent = 0.0f;
    unsigned ballot = __builtin_amdgcn_ballot_w32(active);
    if (ballot != 0u) {   // wave-uniform branch: no divergence, loads stay coalesced
        const int x = x0 + tid;
        const float* p = pred + (((long long)b * C) * H + y) * W + x;
        float v[C];
#pragma unroll
        for (int cix = 0; cix < C; ++cix)
            v[cix] = p[(long long)cix * H * W];
        float m = v[0];
#pragma unroll
        for (int cix = 1; cix < C; ++cix) m = fmaxf(m, v[cix]);
        float s = 0.0f;
#pragma unroll
        for (int cix = 0; cix < C; ++cix) { v[cix] = expf(v[cix] - m); s += v[cix]; }
        const float inv = 1.0f / s;
        float hacc = 0.0f;
#pragma unroll
        for (int cix = 0; cix < C; ++cix) {
            float pr = v[cix] * inv;
            hacc -= pr * logf(pr + 1e-8f);
        }
        ent = active ? hacc : 0.0f;
    }

    // ---- block reduction: WMMA wave sum, then 8 wave partials via LDS ----
    float ws = wave_sum_wmma(ent);        // EXEC all-ones here (no divergence)
    const int wave = tid >> 5;
    if ((tid & 31) == 0) wsum[wave] = ws;
    __syncthreads();
    if (tid == 0) {
        float t = 0.0f;
#pragma unroll
        for (int wv = 0; wv < TILE / 32; ++wv) t += wsum[wv];
        partial[(blockIdx.z * gridDim.y + blockIdx.y) * gridDim.x + blockIdx.x] = t;
    }
}

__global__ __launch_bounds__(256)
void final_reduce(const float* __restrict__ partial, int n, float* __restrict__ out) {
    __shared__ float sm[256];
    float s = 0.0f;
    for (int i = threadIdx.x; i < n; i += 256) s += partial[i];  // fixed order
    sm[threadIdx.x] = s;
    __syncthreads();
    for (int off = 128; off >= 1; off >>= 1) {
        if ((int)threadIdx.x < off) sm[threadIdx.x] += sm[threadIdx.x + off];
        __syncthreads();
    }
    if (threadIdx.x == 0)
        out[0] = sm[0] * (1.0f / ((float)B * (float)H * (float)W));
}

extern "C" void kernel_launch(void* const* d_in, const int* in_sizes, int n_in,
                              void* d_out, int out_size, void* d_ws, size_t ws_size,
                              hipStream_t stream) {
    (void)in_sizes; (void)n_in; (void)out_size; (void)ws_size;
    const float*     pred   = (const float*)d_in[0];      // [8,19,512,512] f32
    const long long* labels = (const long long*)d_in[1];  // [8,512,512] int64
    float* out     = (float*)d_out;                       // scalar f32
    float* partial = (float*)d_ws;                        // 8192 floats = 32 KB

    dim3 grid(W / TILE, H, B);   // 2 x 512 x 8 = 8192 blocks, one 256-wide strip each
    boundary_entropy_partial<<<grid, TILE, 0, stream>>>(pred, labels, partial);

    const int nblocks = (W / TILE) * H * B;
    final_reduce<<<1, 256, 0, stream>>>(partial, nblocks, out);
}